// MapleAttention_49074296324448
// MI455X (gfx1250) — compile-verified
//
#include <hip/hip_runtime.h>
#include <hip/hip_bf16.h>
#include <math.h>

#define B_   64
#define S_   2048
#define H_   2048
#define HQ_  16
#define HKV_ 4
#define D_   128
#define RD_  64
#define G_   (HQ_ / HKV_)
#define EPS_ 1e-6f
#define SCALE_ 0.08838834764831845f   // 128^-0.5

typedef __attribute__((ext_vector_type(2))) float v2f;
typedef __attribute__((ext_vector_type(8))) float v8f;

#define WMMA_F32(A, B, C) \
  __builtin_amdgcn_wmma_f32_16x16x4_f32(false, (A), false, (B), (short)0, (C), false, false)

// ---------------------------------------------------------------------------
// Y[M x N] = X[M x K] * W[N x K]^T using V_WMMA_F32_16X16X4_F32.
// One wave computes TWO adjacent 16x16 N-tiles (shared A fragment), with a
// one-iteration software-pipelined prefetch so loads for K-block k+8 are in
// flight while the WMMAs for K-block k execute (avoids s_wait_loadcnt 0 in
// front of every WMMA). Four accumulators -> four independent WMMA chains.
//
// Fragment layouts (ISA 7.12.2, 32-bit):
//   A 16x4 : lane<16 row M=lane, VGPR0=K0,VGPR1=K1; lane>=16 same row, K2/K3.
//   B 4x16 : mirror (lane half selects K pair).
//   C/D    : VGPR r -> row r + 8*(lane>=16), col = lane&15.
// ---------------------------------------------------------------------------
__global__ __launch_bounds__(32)
void gemm_xwt_wmma(const float* __restrict__ X,
                   const float* __restrict__ W,
                   float* __restrict__ Y,
                   int K, int N) {
  const int tn   = blockIdx.x * 2;   // first of two N tiles
  const int tm   = blockIdx.y;       // M tile
  const int lane = threadIdx.x;      // 0..31
  const int half = lane >> 4;        // 0/1 -> K sub-pair
  const int l15  = lane & 15;
  const int kh2  = half * 2;

  const float* xrow  = X + (long)(tm * 16 + l15) * K + kh2;   // A row m = l15
  const float* wrow0 = W + (long)(tn * 16 + l15) * K + kh2;   // B tile 0
  const float* wrow1 = wrow0 + 16L * K;                       // B tile 1

  v8f c0a = {}, c0b = {}, c1a = {}, c1b = {};

  // prologue: first K-block (8 K values = 2 WMMA steps) in registers
  v2f a0  = *(const v2f*)(xrow);
  v2f a1  = *(const v2f*)(xrow + 4);
  v2f b00 = *(const v2f*)(wrow0);
  v2f b01 = *(const v2f*)(wrow0 + 4);
  v2f b10 = *(const v2f*)(wrow1);
  v2f b11 = *(const v2f*)(wrow1 + 4);

  int k = 0;
  for (; k < K - 8; k += 8) {
    // prefetch next K-block before consuming the current one
    const int kn = k + 8;
    v2f na0  = *(const v2f*)(xrow + kn);
    v2f na1  = *(const v2f*)(xrow + kn + 4);
    v2f nb00 = *(const v2f*)(wrow0 + kn);
    v2f nb01 = *(const v2f*)(wrow0 + kn + 4);
    v2f nb10 = *(const v2f*)(wrow1 + kn);
    v2f nb11 = *(const v2f*)(wrow1 + kn + 4);

    c0a = WMMA_F32(a0, b00, c0a);
    c1a = WMMA_F32(a0, b10, c1a);
    c0b = WMMA_F32(a1, b01, c0b);
    c1b = WMMA_F32(a1, b11, c1b);

    a0 = na0; a1 = na1;
    b00 = nb00; b01 = nb01;
    b10 = nb10; b11 = nb11;
  }
  // epilogue: last K-block
  c0a = WMMA_F32(a0, b00, c0a);
  c1a = WMMA_F32(a0, b10, c1a);
  c0b = WMMA_F32(a1, b01, c0b);
  c1b = WMMA_F32(a1, b11, c1b);

  const v8f c0 = c0a + c0b;
  const v8f c1 = c1a + c1b;

  const int col0 = tn * 16 + l15;
  const int col1 = col0 + 16;
#pragma unroll
  for (int r = 0; r < 8; ++r) {
    const int row = tm * 16 + r + 8 * half;
    Y[(long)row * N + col0] = c0[r];
    Y[(long)row * N + col1] = c1[r];
  }
}

// ---------------------------------------------------------------------------
// Fused RMSNorm + RoPE, in place on q_ws / k_ws.
// grid = (B, HQ+HKV); block = 128 (one thread per dim element).
// ---------------------------------------------------------------------------
__global__ __launch_bounds__(D_)
void norm_rope(float* __restrict__ q_ws,          // [B, HQ*D]
               float* __restrict__ k_ws,          // [B, HKV*D]
               const float* __restrict__ q_norm_w,
               const float* __restrict__ k_norm_w,
               const float* __restrict__ cosb,    // [B, RD]
               const float* __restrict__ sinb) {  // [B, RD]
  const int b    = blockIdx.x;
  const int hidx = blockIdx.y;       // 0..15 = q heads, 16..19 = k heads
  const int d    = threadIdx.x;      // 0..127
  const bool is_q = hidx < HQ_;

  float* base = is_q ? (q_ws + ((long)b * HQ_ + hidx) * D_)
                     : (k_ws + ((long)b * HKV_ + (hidx - HQ_)) * D_);
  const float* w = is_q ? q_norm_w : k_norm_w;

  __shared__ float red[D_];
  __shared__ float xn[D_];

  const float x = base[d];
  red[d] = x * x;
  __syncthreads();
#pragma unroll
  for (int off = D_ / 2; off > 0; off >>= 1) {
    if (d < off) red[d] += red[d + off];
    __syncthreads();
  }
  const float inv = rsqrtf(red[0] / (float)D_ + EPS_);
  const float v = w[d] * x * inv;
  xn[d] = v;
  __syncthreads();

  float out = v;
  if (d < RD_) {
    const float rh = (d < RD_ / 2) ? -xn[d + RD_ / 2] : xn[d - RD_ / 2];
    out = v * cosb[b * RD_ + d] + rh * sinb[b * RD_ + d];
  }
  base[d] = out;
}

// ---------------------------------------------------------------------------
// Flash-decode attention. grid = (B, HKV); block = 256 (8 waves).
// The "new" K/V row (the one the reference appends at out_cache_loc[b]) is
// read from workspace instead of the cache, so d_in is never mutated.
// ---------------------------------------------------------------------------
__global__ __launch_bounds__(256)
void attn_decode(const float* __restrict__ q_ws,        // [B, HQ, D]
                 const float* __restrict__ k_ws,        // [B, HKV, D]
                 const float* __restrict__ v_ws,        // [B, HKV, D]
                 const float* __restrict__ key_cache,   // [P, HKV, D]
                 const float* __restrict__ value_cache, // [P, HKV, D]
                 const int* __restrict__ req_to_token,  // [B, S]
                 const int* __restrict__ req_pool_indices,
                 const int* __restrict__ seq_lens,
                 const int* __restrict__ out_cache_loc,
                 float* __restrict__ attn_out) {        // [B, HQ*D]
  const int b   = blockIdx.x;
  const int kh  = blockIdx.y;
  const int tid = threadIdx.x;                 // 0..255
  const int L   = seq_lens[b];
  const int rp  = req_pool_indices[b];
  const int loc = out_cache_loc[b];
  const int* tok_row = req_to_token + (long)rp * S_;

  const float* knew = k_ws + ((long)b * HKV_ + kh) * D_;
  const float* vnew = v_ws + ((long)b * HKV_ + kh) * D_;

  __shared__ float sc[G_][S_];          // 32 KB scores/probs
  __shared__ float qs[G_][D_];          // q group
  __shared__ float red2[256];
  __shared__ float ssum[G_];
  __shared__ float part[2][G_][D_];     // seq-half partials for P*V

  // load q group (4 heads x 128) into LDS
  for (int i = tid; i < G_ * D_; i += 256)
    qs[i / D_][i % D_] = q_ws[((long)b * HQ_ + kh * G_) * D_ + i];
  __syncthreads();

  // ---- phase 1: scores = (q . k) * scale, lanes along seq ----------------
  for (int s = tid; s < L; s += 256) {
    const int tok = tok_row[s];
    const float* krow = (tok == loc) ? knew
                                     : (key_cache + ((long)tok * HKV_ + kh) * D_);
    float a0 = 0.f, a1 = 0.f, a2 = 0.f, a3 = 0.f;
#pragma unroll 4
    for (int d0 = 0; d0 < D_; d0 += 4) {
      const float4 kv = *(const float4*)(krow + d0);
      const float4 q0 = *(const float4*)(&qs[0][d0]);
      const float4 q1 = *(const float4*)(&qs[1][d0]);
      const float4 q2 = *(const float4*)(&qs[2][d0]);
      const float4 q3 = *(const float4*)(&qs[3][d0]);
      a0 += kv.x * q0.x + kv.y * q0.y + kv.z * q0.z + kv.w * q0.w;
      a1 += kv.x * q1.x + kv.y * q1.y + kv.z * q1.z + kv.w * q1.w;
      a2 += kv.x * q2.x + kv.y * q2.y + kv.z * q2.z + kv.w * q2.w;
      a3 += kv.x * q3.x + kv.y * q3.y + kv.z * q3.z + kv.w * q3.w;
    }
    sc[0][s] = a0 * SCALE_;
    sc[1][s] = a1 * SCALE_;
    sc[2][s] = a2 * SCALE_;
    sc[3][s] = a3 * SCALE_;
  }
  __syncthreads();

  // ---- phase 2: softmax per head (scores -> probs, keep sum) -------------
  for (int h = 0; h < G_; ++h) {
    float m = -INFINITY;
    for (int s = tid; s < L; s += 256) m = fmaxf(m, sc[h][s]);
    red2[tid] = m;
    __syncthreads();
    for (int off = 128; off > 0; off >>= 1) {
      if (tid < off) red2[tid] = fmaxf(red2[tid], red2[tid + off]);
      __syncthreads();
    }
    const float mx = red2[0];
    __syncthreads();

    float sum = 0.f;
    for (int s = tid; s < L; s += 256) {
      const float e = __expf(sc[h][s] - mx);
      sc[h][s] = e;
      sum += e;
    }
    red2[tid] = sum;
    __syncthreads();
    for (int off = 128; off > 0; off >>= 1) {
      if (tid < off) red2[tid] += red2[tid + off];
      __syncthreads();
    }
    if (tid == 0) ssum[h] = red2[0];
    __syncthreads();
  }

  // ---- phase 3: out = P * V, lanes along D (coalesced V rows) ------------
  const int d   = tid & (D_ - 1);
  const int grp = tid >> 7;             // 0/1 -> even/odd seq positions
  float a0 = 0.f, a1 = 0.f, a2 = 0.f, a3 = 0.f;
  for (int s = grp; s < L; s += 2) {
    const int tok = tok_row[s];
    const float* vrow = (tok == loc) ? vnew
                                     : (value_cache + ((long)tok * HKV_ + kh) * D_);
    const float vv = vrow[d];
    a0 += sc[0][s] * vv;
    a1 += sc[1][s] * vv;
    a2 += sc[2][s] * vv;
    a3 += sc[3][s] * vv;
  }
  part[grp][0][d] = a0;
  part[grp][1][d] = a1;
  part[grp][2][d] = a2;
  part[grp][3][d] = a3;
  __syncthreads();

  if (grp == 0) {
#pragma unroll
    for (int h = 0; h < G_; ++h) {
      const float o = (part[0][h][d] + part[1][h][d]) / ssum[h];
      attn_out[(long)b * (HQ_ * D_) + (kh * G_ + h) * D_ + d] = o;
    }
  }
}

// ---------------------------------------------------------------------------
extern "C" void kernel_launch(void* const* d_in, const int* in_sizes, int n_in,
                              void* d_out, int out_size, void* d_ws, size_t ws_size,
                              hipStream_t stream) {
  (void)in_sizes; (void)n_in; (void)out_size; (void)ws_size;

  const float* hidden = (const float*)d_in[0];
  const float* Wq     = (const float*)d_in[1];
  const float* Wk     = (const float*)d_in[2];
  const float* Wv     = (const float*)d_in[3];
  const float* qnw    = (const float*)d_in[4];
  const float* knw    = (const float*)d_in[5];
  const float* Wo     = (const float*)d_in[6];
  const float* kc     = (const float*)d_in[7];
  const float* vc     = (const float*)d_in[8];
  const float* cosb   = (const float*)d_in[9];
  const float* sinb   = (const float*)d_in[10];
  const int*   r2t    = (const int*)d_in[11];
  const int*   rpi    = (const int*)d_in[12];
  const int*   slen   = (const int*)d_in[13];
  const int*   ocl    = (const int*)d_in[14];
  float* out = (float*)d_out;

  float* q_ws = (float*)d_ws;                  // 64*2048 f32
  float* k_ws = q_ws + B_ * HQ_ * D_;          // 64*512
  float* v_ws = k_ws + B_ * HKV_ * D_;         // 64*512
  float* attn = v_ws + B_ * HKV_ * D_;         // 64*2048

  const dim3 wblk(32);
  // QKV projections (WMMA f32): Y = X * W^T   (grid.x = N/32: 2 tiles/wave)
  gemm_xwt_wmma<<<dim3((HQ_ * D_) / 32, B_ / 16), wblk, 0, stream>>>(hidden, Wq, q_ws, H_, HQ_ * D_);
  gemm_xwt_wmma<<<dim3((HKV_ * D_) / 32, B_ / 16), wblk, 0, stream>>>(hidden, Wk, k_ws, H_, HKV_ * D_);
  gemm_xwt_wmma<<<dim3((HKV_ * D_) / 32, B_ / 16), wblk, 0, stream>>>(hidden, Wv, v_ws, H_, HKV_ * D_);

  // RMSNorm + RoPE (q: 16 heads, k: 4 heads)
  norm_rope<<<dim3(B_, HQ_ + HKV_), dim3(D_), 0, stream>>>(q_ws, k_ws, qnw, knw, cosb, sinb);

  // Flash-decode attention per (batch, kv head)
  attn_decode<<<dim3(B_, HKV_), dim3(256), 0, stream>>>(q_ws, k_ws, v_ws, kc, vc,
                                                        r2t, rpi, slen, ocl, attn);

  // Output projection (WMMA f32) straight into d_out
  gemm_xwt_wmma<<<dim3(H_ / 32, B_ / 16), wblk, 0, stream>>>(attn, Wo, out, HQ_ * D_, H_);
}